// GCN_69097433858168
// MI455X (gfx1250) — compile-verified
//
#include <hip/hip_runtime.h>

#define NN 100000
#define NE 1200000
#define NG 128
#define DF 64
#define NC 10

typedef __attribute__((ext_vector_type(2))) float v2f;
typedef __attribute__((ext_vector_type(8))) float v8f;

// ---------------- degree / normalization ----------------
__global__ void k_init_deg(float* __restrict__ deg) {
    int i = blockIdx.x * blockDim.x + threadIdx.x;
    if (i < NN) deg[i] = 1.0f;   // self-loop contributes 1
}

__global__ void k_edge_deg(const int* __restrict__ col, float* __restrict__ deg) {
    int e = blockIdx.x * blockDim.x + threadIdx.x;
    if (e < NE) atomicAdd(&deg[col[e]], 1.0f);
}

__global__ void k_dinv(float* __restrict__ deg) {
    int i = blockIdx.x * blockDim.x + threadIdx.x;
    if (i < NN) deg[i] = rsqrtf(fmaxf(deg[i], 1.0f));
}

// ---------------- dense transform: C[N,64] = A[N,64] @ W[64,64] ----------------
// One wave32 computes one 16x16 output tile with V_WMMA_F32_16X16X4_F32,
// chaining the f32 accumulator over 16 K-chunks of 4.
// W (16 KB) is staged once per block into LDS via the CDNA5 async global->LDS
// path (GLOBAL_LOAD_ASYNC_TO_LDS_B128, ASYNCcnt), then B operands are read
// from LDS; A operands stream from global as contiguous b64 loads.
// A layout (16x4 f32): lane = M (lane&15); VGPR v holds K = 2*(lane>>4) + v.
// B layout (4x16 f32): lane = N (lane&15); VGPR v holds K = 2*(lane>>4) + v.
// C/D layout (16x16 f32, 8 VGPRs): lane = N; VGPR v holds M = v + 8*(lane>>4).
__global__ void k_gemm64_wmma(const float* __restrict__ A,
                              const float* __restrict__ W,
                              float* __restrict__ C) {
    __shared__ float sW[DF * DF];   // 16 KB: full 64x64 W, row-major

    // Async-stage W: 256 threads x 16 bytes x 4 rounds = 16 KB (GVS addressing:
    // LDS byte offset in VGPR, 32-bit global offset in VGPR, base in SGPR pair).
    {
        unsigned t = threadIdx.x;
        unsigned lds0 = (unsigned)(uintptr_t)&sW[0];
#pragma unroll
        for (int i = 0; i < 4; ++i) {
            unsigned off = t * 16u + (unsigned)i * 4096u;
            unsigned ldsaddr = lds0 + off;
            asm volatile("global_load_async_to_lds_b128 %0, %1, %2"
                         :: "v"(ldsaddr), "v"(off), "s"(W) : "memory");
        }
        asm volatile("s_wait_asynccnt 0" ::: "memory");
    }
    __syncthreads();

    int wave = (blockIdx.x * blockDim.x + threadIdx.x) >> 5;   // global wave id = tile id
    int lane = threadIdx.x & 31;
    int rowTile = wave >> 2;      // 0..6249   (100000 = 6250*16, exact)
    int colTile = wave & 3;       // 0..3      (64 = 4*16, exact)
    int half = lane >> 4;         // 0 or 1
    int mn   = lane & 15;

    const float* Arow = A + (size_t)(rowTile * 16 + mn) * DF;
    const float* sWcol = sW + colTile * 16 + mn;

    v8f c = {0.f, 0.f, 0.f, 0.f, 0.f, 0.f, 0.f, 0.f};
#pragma unroll
    for (int kc = 0; kc < 16; ++kc) {
        int k = kc * 4 + half * 2;
        v2f a = *(const v2f*)(Arow + k);          // A[m][k], A[m][k+1] contiguous
        v2f b;
        b.x = sWcol[k * DF];                      // W[k][n]   (LDS)
        b.y = sWcol[(k + 1) * DF];                // W[k+1][n] (LDS)
        c = __builtin_amdgcn_wmma_f32_16x16x4_f32(
                /*neg_a=*/false, a, /*neg_b=*/false, b,
                /*c_mod=*/(short)0, c, /*reuse_a=*/false, /*reuse_b=*/false);
    }

    float* Cout = C + (size_t)(rowTile * 16 + half * 8) * DF + colTile * 16 + mn;
#pragma unroll
    for (int v = 0; v < 8; ++v) Cout[(size_t)v * DF] = c[v];
}

// ---------------- self-loop term (also fully initializes H1) ----------------
__global__ void k_selfloop(const float* __restrict__ H0, const float* __restrict__ dinv,
                           float* __restrict__ H1) {
    int i = blockIdx.x * blockDim.x + threadIdx.x;
    if (i < NN * DF) {
        float d = dinv[i >> 6];
        H1[i] = H0[i] * d * d;    // norm for self-loop edge (i,i) = dinv[i]^2
    }
}

// ---------------- edge scatter: H1[col] += H0[row] * dinv[row]*dinv[col] ----------------
// 16 threads per edge, float4 gather, 4 f32 atomics each. Whole feature matrix is
// L2-resident (25.6MB << 192MB L2), so the atomics resolve on-chip.
__global__ void k_scatter(const float* __restrict__ H0, const float* __restrict__ dinv,
                          const int* __restrict__ row, const int* __restrict__ col,
                          float* __restrict__ H1) {
    long long t = (long long)blockIdx.x * blockDim.x + threadIdx.x;
    int e  = (int)(t >> 4);
    int ch = (int)(t & 15);
    if (e < NE) {
        int r = row[e], cd = col[e];
        float w = dinv[r] * dinv[cd];
        const float4 v = *(const float4*)(H0 + (size_t)r * DF + ch * 4);
        float* dst = H1 + (size_t)cd * DF + ch * 4;
        atomicAdd(dst + 0, v.x * w);
        atomicAdd(dst + 1, v.y * w);
        atomicAdd(dst + 2, v.z * w);
        atomicAdd(dst + 3, v.w * w);
    }
}

// ---------------- bias + optional ReLU, in place ----------------
__global__ void k_bias_act(float* __restrict__ H, const float* __restrict__ b, int relu) {
    int i = blockIdx.x * blockDim.x + threadIdx.x;
    if (i < NN * DF) {
        float v = H[i] + b[i & 63];
        if (relu) v = fmaxf(v, 0.0f);
        H[i] = v;
    }
}

// ---------------- pooling ----------------
__global__ void k_zero(float* __restrict__ p, int n) {
    int i = blockIdx.x * blockDim.x + threadIdx.x;
    if (i < n) p[i] = 0.0f;
}

__global__ void k_pool(const float* __restrict__ H, const int* __restrict__ batch,
                       float* __restrict__ g, float* __restrict__ cnt) {
    int i = blockIdx.x * blockDim.x + threadIdx.x;
    if (i < NN * DF) {
        int n = i >> 6, f = i & 63;
        int bi = batch[n];
        atomicAdd(&g[bi * DF + f], H[i]);
        if (f == 0) atomicAdd(&cnt[bi], 1.0f);
    }
}

__global__ void k_final(const float* __restrict__ g, const float* __restrict__ cnt,
                        const float* __restrict__ linW, const float* __restrict__ linb,
                        float* __restrict__ out) {
    int t = blockIdx.x * blockDim.x + threadIdx.x;
    if (t < NG * NC) {
        int bg = t / NC, cl = t % NC;
        float inv = 1.0f / fmaxf(cnt[bg], 1.0f);
        float s = linb[cl];
#pragma unroll 8
        for (int f = 0; f < DF; ++f)
            s += g[bg * DF + f] * inv * linW[f * NC + cl];
        out[t] = s;
    }
}

// ---------------- launch ----------------
extern "C" void kernel_launch(void* const* d_in, const int* in_sizes, int n_in,
                              void* d_out, int out_size, void* d_ws, size_t ws_size,
                              hipStream_t stream) {
    (void)in_sizes; (void)n_in; (void)out_size; (void)ws_size;

    const float* x     = (const float*)d_in[0];
    const int*   eidx  = (const int*)d_in[1];   // [2, NE] flattened
    const int*   batch = (const int*)d_in[2];
    const float* W1 = (const float*)d_in[3];
    const float* b1 = (const float*)d_in[4];
    const float* W2 = (const float*)d_in[5];
    const float* b2 = (const float*)d_in[6];
    const float* W3 = (const float*)d_in[7];
    const float* b3 = (const float*)d_in[8];
    const float* linW = (const float*)d_in[9];
    const float* linb = (const float*)d_in[10];
    const int* row = eidx;
    const int* col = eidx + NE;

    float* ws   = (float*)d_ws;
    float* dinv = ws;                         // NN
    float* H0   = dinv + NN;                  // NN*DF
    float* H1   = H0 + (size_t)NN * DF;       // NN*DF
    float* g    = H1 + (size_t)NN * DF;       // NG*DF
    float* cnt  = g + NG * DF;                // NG (contiguous after g)

    // symmetric normalization, computed once, reused for all 3 layers
    k_init_deg<<<(NN + 255) / 256, 256, 0, stream>>>(dinv);
    k_edge_deg<<<(NE + 255) / 256, 256, 0, stream>>>(col, dinv);
    k_dinv<<<(NN + 255) / 256, 256, 0, stream>>>(dinv);

    const float* Ws[3] = {W1, W2, W3};
    const float* bs[3] = {b1, b2, b3};
    const float* in = x;
    for (int l = 0; l < 3; ++l) {
        // 25000 tiles, 8 waves/block -> exactly 3125 blocks
        k_gemm64_wmma<<<3125, 256, 0, stream>>>(in, Ws[l], H0);
        k_selfloop<<<(NN * DF + 255) / 256, 256, 0, stream>>>(H0, dinv, H1);
        k_scatter<<<75000, 256, 0, stream>>>(H0, dinv, row, col, H1);  // NE*16 threads exact
        k_bias_act<<<(NN * DF + 255) / 256, 256, 0, stream>>>(H1, bs[l], l < 2 ? 1 : 0);
        in = H1;
    }

    k_zero<<<(NG * DF + NG + 255) / 256, 256, 0, stream>>>(g, NG * DF + NG); // g + cnt
    k_pool<<<(NN * DF + 255) / 256, 256, 0, stream>>>(H1, batch, g, cnt);
    k_final<<<(NG * NC + 255) / 256, 256, 0, stream>>>(g, cnt, linW, linb, (float*)d_out);
}